// Yolov3_82746839925232
// MI455X (gfx1250) — compile-verified
//
#include <hip/hip_runtime.h>
#include <cstdint>
#include <cstddef>

// ---------------------------------------------------------------------------
// YOLOv3 focal + IoU loss (B=8, C=80, A=68229, M=100) as a single fused
// streaming-reduction pass.
//
// Roofline: cls_dt is 174.7 MB read once -> ~7.5us floor at 23.3 TB/s HBM.
// Per element: ~6 VALU + 1 v_log_f32 (focal loss algebraically split so each
// element needs one log; the lone positive-class element per anchor is fixed
// up after the class loop). Compute sits under the bandwidth envelope, so the
// kernel is built purely for full-bandwidth streaming.
//
// CDNA5 path: cls_dt tiles are staged via GLOBAL_LOAD_ASYNC_TO_LDS_B32
// (ASYNCcnt-tracked), double-buffered in LDS, pipelined with
// s_wait_asynccnt<=8 so the next tile's loads are in flight while the current
// tile is consumed. b32 granularity because row stride A=68229 floats makes
// row bases only 4B-aligned (b128 would be misaligned).
// ---------------------------------------------------------------------------

#define A_N 68229
#define B_N 8
#define C_N 80
#define M_N 100
#define ALPHA_F 0.25f
#define EPS_F 1e-7f

#define BLK 256
#define TILE_C 8
#define NTILE (C_N / TILE_C)             // 10 tiles of 8 classes
#define NBLK ((A_N + BLK - 1) / BLK)     // 267 anchor chunks

#if defined(__AMDGCN__) && __has_builtin(__builtin_amdgcn_global_load_async_to_lds_b32)
#define USE_ASYNC_LDS 1
#endif

__device__ __forceinline__ void async_copy_b32(const float* g, float* l) {
#ifdef USE_ASYNC_LDS
  // Builtin signature (from clang diagnostic): (__device__ int*, __local int*,
  // imm offset, imm cpol). AS1 pointer built via integer (numerically the
  // global VA); AS3 pointer is the low 32 bits of the generic LDS address
  // (aperture form: addr[31:0] = LDS offset, CDNA5 ISA 10.2).
  __builtin_amdgcn_global_load_async_to_lds_b32(
      (__attribute__((address_space(1))) int*)(uintptr_t)g,
      (__attribute__((address_space(3))) int*)(uint32_t)(uintptr_t)l,
      0, 0);
#else
  *l = *g;  // fallback: synchronous VGPR-staged copy (same LDS layout)
#endif
}

__device__ __forceinline__ void wait_async_tile() {
#ifdef USE_ASYNC_LDS
#if __has_builtin(__builtin_amdgcn_s_wait_asynccnt)
  __builtin_amdgcn_s_wait_asynccnt(TILE_C);   // allow next tile's 8 loads in flight
#else
  asm volatile("s_wait_asynccnt 8" ::: "memory");
#endif
#endif
}

__device__ __forceinline__ void wait_async_zero() {
#ifdef USE_ASYNC_LDS
#if __has_builtin(__builtin_amdgcn_s_wait_asynccnt)
  __builtin_amdgcn_s_wait_asynccnt(0);
#else
  asm volatile("s_wait_asynccnt 0" ::: "memory");
#endif
#endif
}

__global__ __launch_bounds__(BLK)
void yolo_loss_main(const float* __restrict__ cls_dt,
                    const float* __restrict__ reg_dt,
                    const int*   __restrict__ assign_result,
                    const float* __restrict__ gt_boxes,
                    const int*   __restrict__ gt_cls,
                    const float* __restrict__ anch_x,
                    const float* __restrict__ anch_y,
                    const float* __restrict__ anch_w,
                    const float* __restrict__ anch_h,
                    float* __restrict__ ws) {
  __shared__ float sbuf[2][TILE_C * BLK];   // 2 x 8KB double buffer
  __shared__ float s_gtb[M_N * 4];          // this batch's gt boxes
  __shared__ int   s_gtc[M_N];              // this batch's gt classes
  __shared__ float s_rc[BLK / 32], s_rb[BLK / 32];
  __shared__ int   s_rn[BLK / 32];

  const int tid = threadIdx.x;
  const int bx  = blockIdx.x;               // anchor chunk
  const int b   = blockIdx.y;               // batch index
  const int a0  = bx * BLK;
  const int a   = a0 + tid;
  const bool act = (a < A_N);
  const int aL  = act ? a : (A_N - 1);      // clamped address, data unused if !act

  const float* clsb = cls_dt + (size_t)b * C_N * A_N;

  // ---- kick off async loads for tile 0 (8 per wave, uniform EXEC) ----
#pragma unroll
  for (int j = 0; j < TILE_C; ++j)
    async_copy_b32(clsb + (size_t)j * A_N + aL, &sbuf[0][j * BLK + tid]);

  // ---- stage gt boxes/classes for this batch in LDS (tiny, 2KB) ----
  for (int i = tid; i < M_N * 4; i += BLK) s_gtb[i] = gt_boxes[(size_t)b * M_N * 4 + i];
  for (int i = tid; i < M_N;     i += BLK) s_gtc[i] = gt_cls[(size_t)b * M_N + i];

  // ---- per-anchor metadata ----
  int assign = act ? (assign_result[(size_t)b * A_N + a] - 1) : -1;
  const bool poscls = assign >= 0;
  const bool posbox = assign >= 1;
  int idx = assign - 1;
  idx = idx < 0 ? 0 : (idx > (M_N - 1) ? (M_N - 1) : idx);

  __syncthreads();                          // gt staging visible
  const int tc = s_gtc[idx];

  // ---- pipelined class-tile loop: sum 0.75*p^2*(-log(1-p)) over all c ----
  float negsum = 0.0f;
  float p_tc   = 0.5f;                      // always overwritten (tc in [0,C))
  for (int t = 0; t < NTILE; ++t) {
    if (t + 1 < NTILE) {
      float* nb = sbuf[(t + 1) & 1];
#pragma unroll
      for (int j = 0; j < TILE_C; ++j)
        async_copy_b32(clsb + (size_t)((t + 1) * TILE_C + j) * A_N + aL,
                       &nb[j * BLK + tid]);
      wait_async_tile();                    // tile t complete, t+1 in flight
    } else {
      wait_async_zero();
    }
    __syncthreads();                        // every wave's tile-t data landed

    const float* buf = sbuf[t & 1];
#pragma unroll
    for (int j = 0; j < TILE_C; ++j) {
      const int c = t * TILE_C + j;
      float p = buf[j * BLK + tid];
      p = fminf(fmaxf(p, EPS_F), 1.0f - EPS_F);
      negsum += (1.0f - ALPHA_F) * p * p * (-__logf(1.0f - p));
      p_tc = (c == tc) ? p : p_tc;
    }
    __syncthreads();                        // done reading before buffer reuse
  }

  // ---- fix up the single positive-class element ----
  float cls = negsum;
  {
    const float p  = p_tc;
    const float nt = (1.0f - ALPHA_F) * p * p * (-__logf(1.0f - p));
    const float pt = ALPHA_F * (1.0f - p) * (1.0f - p) * (-__logf(p));
    if (posbox) cls += pt - nt;
  }
  if (!(act && poscls)) cls = 0.0f;

  // ---- box branch: decode + IoU ----
  const float ax = anch_x[aL], ay = anch_y[aL];
  const float aw = anch_w[aL], ah = anch_h[aL];
  const float* regb = reg_dt + (size_t)b * 4 * A_N;
  const float r0 = regb[aL];
  const float r1 = regb[(size_t)A_N + aL];
  const float r2 = regb[(size_t)2 * A_N + aL];
  const float r3 = regb[(size_t)3 * A_N + aL];
  const float dx = ax + r0 * aw;
  const float dy = ay + r1 * ah;
  const float dw = aw * __expf(fminf(r2, 50.0f));
  const float dh = ah * __expf(fminf(r3, 50.0f));
  const float g0 = s_gtb[idx * 4 + 0], g1 = s_gtb[idx * 4 + 1];
  const float g2 = s_gtb[idx * 4 + 2], g3 = s_gtb[idx * 4 + 3];
  const float tx = g0 + 0.5f * g2, ty = g1 + 0.5f * g3;
  const float ix1 = fmaxf(dx - 0.5f * dw, tx - 0.5f * g2);
  const float iy1 = fmaxf(dy - 0.5f * dh, ty - 0.5f * g3);
  const float ix2 = fminf(dx + 0.5f * dw, tx + 0.5f * g2);
  const float iy2 = fminf(dy + 0.5f * dh, ty + 0.5f * g3);
  const float inter = fmaxf(ix2 - ix1, 0.0f) * fmaxf(iy2 - iy1, 0.0f);
  const float uni   = dw * dh + g2 * g3 - inter;
  const float iou   = inter / (uni + EPS_F);
  float boxl = (act && posbox) ? (1.0f - iou) : 0.0f;
  int   cnt  = (act && posbox) ? 1 : 0;

  // ---- block reduction: wave32 shuffles then 8 per-wave partials ----
#pragma unroll
  for (int o = 16; o > 0; o >>= 1) {
    cls  += __shfl_down(cls,  o, 32);
    boxl += __shfl_down(boxl, o, 32);
    cnt  += __shfl_down(cnt,  o, 32);
  }
  const int wv = tid >> 5;
  if ((tid & 31) == 0) { s_rc[wv] = cls; s_rb[wv] = boxl; s_rn[wv] = cnt; }
  __syncthreads();
  if (tid == 0) {
    float sc = 0.0f, sx = 0.0f; int sn = 0;
#pragma unroll
    for (int w = 0; w < BLK / 32; ++w) { sc += s_rc[w]; sx += s_rb[w]; sn += s_rn[w]; }
    float* o = ws + ((size_t)b * NBLK + bx) * 3;
    o[0] = sc; o[1] = sx; o[2] = (float)sn;  // overwrite (not accumulate): deterministic
  }
}

// Deterministic finalize: fixed-order per-batch sums, then /num_pos and /B.
__global__ __launch_bounds__(32)
void yolo_finalize(const float* __restrict__ ws, float* __restrict__ out) {
  __shared__ float acc[B_N];
  const int b = threadIdx.x;
  if (b < B_N) {
    float c = 0.0f, x = 0.0f, n = 0.0f;
    for (int i = 0; i < NBLK; ++i) {
      const float* p = ws + ((size_t)b * NBLK + i) * 3;
      c += p[0]; x += p[1]; n += p[2];
    }
    const float np = fmaxf(n, 1.0f);
    acc[b] = (c + x) / np;
  }
  __syncthreads();
  if (threadIdx.x == 0) {
    float s = 0.0f;
#pragma unroll
    for (int i = 0; i < B_N; ++i) s += acc[i];
    out[0] = s / (float)B_N;
  }
}

extern "C" void kernel_launch(void* const* d_in, const int* in_sizes, int n_in,
                              void* d_out, int out_size, void* d_ws, size_t ws_size,
                              hipStream_t stream) {
  const float* cls_dt   = (const float*)d_in[0];
  const float* reg_dt   = (const float*)d_in[1];
  const int*   assign   = (const int*)d_in[2];
  const float* gt_boxes = (const float*)d_in[3];
  const int*   gt_cls   = (const int*)d_in[4];
  const float* anch_x   = (const float*)d_in[5];
  const float* anch_y   = (const float*)d_in[6];
  const float* anch_w   = (const float*)d_in[7];
  const float* anch_h   = (const float*)d_in[8];
  float* ws = (float*)d_ws;   // needs B_N*NBLK*3 floats = 25.6 KB

  dim3 grid(NBLK, B_N);
  yolo_loss_main<<<grid, dim3(BLK), 0, stream>>>(
      cls_dt, reg_dt, assign, gt_boxes, gt_cls,
      anch_x, anch_y, anch_w, anch_h, ws);
  yolo_finalize<<<dim3(1), dim3(32), 0, stream>>>((const float*)ws, (float*)d_out);
}